// FrechetDistanceLoss_59296318488927
// MI455X (gfx1250) — compile-verified
//
#include <hip/hip_runtime.h>
#include <math.h>

typedef __attribute__((ext_vector_type(2))) float v2f;
typedef __attribute__((ext_vector_type(8))) float v8f;

#define TT 1024   // sequence length T
#define DD 64     // feature dim
#define BB 32     // batch
#define RR 4      // DP tile size (RR x RR per thread per step)
#define NB (TT / RR)   // 256 block rows/cols

// ---------------------------------------------------------------------------
// Kernel 1: per-point squared norms for pred (a2) and targ (b2).
// ---------------------------------------------------------------------------
__global__ void frechet_norms_kernel(const float* __restrict__ pred,
                                     const float* __restrict__ targ,
                                     float* __restrict__ a2,
                                     float* __restrict__ b2) {
    int idx = blockIdx.x * blockDim.x + threadIdx.x;   // 0 .. 2*B*T-1
    const float* src;
    float* dst;
    int p;
    if (idx < BB * TT) { src = pred; dst = a2; p = idx; }
    else               { src = targ; dst = b2; p = idx - BB * TT; }
    const float4* row = (const float4*)(src + (size_t)p * DD);
    float s = 0.0f;
#pragma unroll
    for (int i = 0; i < DD / 4; ++i) {
        float4 v = row[i];
        s += v.x * v.x + v.y * v.y + v.z * v.z + v.w * v.w;
    }
    dst[p] = s;
}

// ---------------------------------------------------------------------------
// Kernel 2: cost[b][i][j] = sqrt(max(a2[i] + b2[j] - 2 * <pred_i, targ_j>, 1e-12))
// GEMM via V_WMMA_F32_16X16X4_F32: one wave per 16x16 output tile, K=64 as
// 16 chained K=4 WMMAs. Block = 256 threads = 8 waves = 8 N-tiles.
// ---------------------------------------------------------------------------
__global__ __launch_bounds__(256)
void frechet_cost_kernel(const float* __restrict__ pred,
                         const float* __restrict__ targ,
                         const float* __restrict__ a2,
                         const float* __restrict__ b2,
                         float* __restrict__ cost) {
    const int lane  = threadIdx.x & 31;
    const int wave  = threadIdx.x >> 5;
    const int tileN = blockIdx.x * 8 + wave;   // 0..63
    const int tileM = blockIdx.y;              // 0..63
    const int b     = blockIdx.z;              // 0..31
    const int g     = lane >> 4;               // lane group 0/1
    const int l     = lane & 15;

    const float* Arow = pred + ((size_t)b * TT + tileM * 16 + l) * DD;
    const float* Brow = targ + ((size_t)b * TT + tileN * 16 + l) * DD;

    v8f acc = {};
#pragma unroll
    for (int kc = 0; kc < DD / 4; ++kc) {
        const int k0 = kc * 4 + 2 * g;         // even -> 8B aligned
        v2f a  = *(const v2f*)(Arow + k0);     // K = k0, k0+1
        v2f bt = *(const v2f*)(Brow + k0);
        acc = __builtin_amdgcn_wmma_f32_16x16x4_f32(
            /*neg_a=*/false, a, /*neg_b=*/false, bt,
            /*c_mod=*/(short)0, acc, /*reuse_a=*/false, /*reuse_b=*/false);
    }

    const int    gn    = tileN * 16 + l;
    const float  bn    = b2[(size_t)b * TT + gn];
    const size_t cbase = (size_t)b * TT * TT;
#pragma unroll
    for (int v = 0; v < 8; ++v) {
        const int   M  = v + 8 * g;
        const int   gm = tileM * 16 + M;
        const float am = a2[(size_t)b * TT + gm];
        const float sq = am + bn - 2.0f * acc[v];
        cost[cbase + (size_t)gm * TT + gn] = sqrtf(fmaxf(sq, 1e-12f));
    }
}

// ---------------------------------------------------------------------------
// Kernel 3: Frechet DP, 4x4-blocked anti-diagonal wavefront.
// 256 threads per batch; thread j owns columns [4j, 4j+4). Block anti-diagonal
// D = I + J, D = 0..510; at step D thread j processes tile (I=D-j, J=j):
// 16 cells serially with float4 cost loads. Cross-thread data: right column
// (4 floats) + corner via two ping-pong LDS buffers; bottom row stays in regs.
// Recurrence: ca[i][j] = max(cost, min(up, left, diag)), OOB neighbors = +INF,
// (0,0) special-cased (reproduces the cummax / first-column boundaries).
// ---------------------------------------------------------------------------
__global__ __launch_bounds__(NB)
void frechet_dp_kernel(const float* __restrict__ cost,
                       float* __restrict__ dists) {
    const int j = threadIdx.x;     // block column 0..255
    const int b = blockIdx.x;
    const float* C = cost + (size_t)b * TT * TT;

    __shared__ float rcol[2][NB][RR];   // right-column ping-pong buffers
    const float INFV = 3.0e38f;

    float up[RR];                       // ca[4I-1][4j..4j+3] (bottom row of tile above)
#pragma unroll
    for (int k = 0; k < RR; ++k) up[k] = INFV;

    int cur = 0;        // rcol[cur] holds D-2 (overwritten with D); rcol[cur^1] holds D-1
    float finalVal = 0.0f;

    for (int D = 0; D < 2 * NB - 1; ++D) {
        const int I = D - j;
        const bool active = (I >= 0) && (I < NB);

        // ---- read phase (from LDS written at steps D-1 / D-2) ----
        float leftCol[RR];
        float corner = INFV;
#pragma unroll
        for (int r = 0; r < RR; ++r) leftCol[r] = INFV;
        if (active && j > 0) {
#pragma unroll
            for (int r = 0; r < RR; ++r) leftCol[r] = rcol[cur ^ 1][j - 1][r];
            if (I > 0) corner = rcol[cur][j - 1][RR - 1];
        }
        __syncthreads();                // reads done before overwrite

        // ---- compute phase: 4x4 tile, row-major ----
        float rc[RR];                   // my right column this tile
        if (active) {
            const int i0 = I * RR;
            const int c0 = j * RR;
            float prev[RR];             // previous in-tile row
#pragma unroll
            for (int r = 0; r < RR; ++r) {
                const float4 c4 = *(const float4*)(C + (size_t)(i0 + r) * TT + c0);
                const float cv[RR] = {c4.x, c4.y, c4.z, c4.w};
                float curRow[RR];
#pragma unroll
                for (int k = 0; k < RR; ++k) {
                    const float up_v   = (r == 0) ? up[k] : prev[k];
                    const float left_v = (k == 0) ? leftCol[r] : curRow[k - 1];
                    const float diag_v = (k == 0)
                        ? ((r == 0) ? corner : leftCol[r - 1])
                        : ((r == 0) ? up[k - 1] : prev[k - 1]);
                    float mn = fminf(fminf(up_v, left_v), diag_v);
                    if (i0 + r == 0 && c0 + k == 0) mn = -INFV;  // ca[0][0] = cost[0][0]
                    curRow[k] = fmaxf(cv[k], mn);
                }
#pragma unroll
                for (int k = 0; k < RR; ++k) prev[k] = curRow[k];
                rc[r] = curRow[RR - 1];
            }
#pragma unroll
            for (int k = 0; k < RR; ++k) up[k] = prev[k];   // bottom row -> regs
            finalVal = rc[RR - 1];                           // tile bottom-right
        }

        // ---- write phase: publish right column into the D-2 buffer ----
        if (active) {
#pragma unroll
            for (int r = 0; r < RR; ++r) rcol[cur][j][r] = rc[r];
        }
        __syncthreads();                // writes visible before next reads
        cur ^= 1;
    }

    if (j == NB - 1) dists[b] = finalVal;   // ca[T-1][T-1]
}

// ---------------------------------------------------------------------------
// Kernel 4: deterministic ordered mean over 32 batch distances.
// ---------------------------------------------------------------------------
__global__ void frechet_reduce_kernel(const float* __restrict__ dists,
                                      float* __restrict__ out) {
    if (threadIdx.x == 0 && blockIdx.x == 0) {
        float s = 0.0f;
        for (int i = 0; i < BB; ++i) s += dists[i];
        out[0] = s * (1.0f / (float)BB);
    }
}

// ---------------------------------------------------------------------------
// Launcher
// ---------------------------------------------------------------------------
extern "C" void kernel_launch(void* const* d_in, const int* in_sizes, int n_in,
                              void* d_out, int out_size, void* d_ws, size_t ws_size,
                              hipStream_t stream) {
    const float* pred = (const float*)d_in[0];   // (32, 1024, 64) f32
    const float* targ = (const float*)d_in[1];   // (32, 1024, 64) f32
    float* out = (float*)d_out;                  // scalar f32

    // Workspace layout:
    //  cost : 32*1024*1024 f32 = 134217728 B (L2-resident on MI455X)
    //  a2   : 32*1024 f32
    //  b2   : 32*1024 f32
    //  dists: 32 f32
    char* ws = (char*)d_ws;
    float* cost  = (float*)ws;
    float* a2    = (float*)(ws + (size_t)BB * TT * TT * sizeof(float));
    float* b2    = a2 + (size_t)BB * TT;
    float* dists = b2 + (size_t)BB * TT;

    // 1) norms
    frechet_norms_kernel<<<(2 * BB * TT) / 256, 256, 0, stream>>>(pred, targ, a2, b2);

    // 2) cost matrices via fp32 WMMA
    dim3 cgrid(8, TT / 16, BB);
    frechet_cost_kernel<<<cgrid, 256, 0, stream>>>(pred, targ, a2, b2, cost);

    // 3) blocked wavefront DP: one 256-thread block per batch
    frechet_dp_kernel<<<BB, NB, 0, stream>>>(cost, dists);

    // 4) ordered mean
    frechet_reduce_kernel<<<1, 1, 0, stream>>>(dists, out);
}